// BayesianLinear_53919019434687
// MI455X (gfx1250) — compile-verified
//
#include <hip/hip_runtime.h>
#include <math.h>

#define INN   4096
#define OUTN  8192
#define BATCH 128
#define NWG   (OUTN / 16)   // 512 single-wave workgroups for the fused kernel

typedef __attribute__((ext_vector_type(2))) float v2f;
typedef __attribute__((ext_vector_type(8))) float v8f;

// math constants
#define LOG2E_F   1.4426950408889634f
#define LN2_F     0.6931471805599453f
#define K1_F     (-0.7213475204444817f)    // -0.5 * log2(e)
#define K2_F     (-115415.60327111707f)    // -0.5 * (1/0.0025)^2 * log2(e)
#define CPRIOR_F (-1.6120857137646196f)    // ln(0.5) - ln(sqrt(2*pi))  per element
#define CPOST_F  (-0.9189385332046727f)    // -ln(sqrt(2*pi))           per element

__device__ __forceinline__ float rcp_fast(float x)  { return __builtin_amdgcn_rcpf(x); }
__device__ __forceinline__ float exp2_raw(float x)  { return __builtin_amdgcn_exp2f(x); }
__device__ __forceinline__ float log2_raw(float x)  { return __builtin_amdgcn_logf(x); }
// softplus(x) = ln(1+e^x) = ln2 * log2(1 + 2^(x*log2e))
__device__ __forceinline__ float softplus_fast(float x) {
    return LN2_F * log2_raw(1.0f + exp2_raw(x * LOG2E_F));
}

// Sample one weight element and accumulate the VARIABLE parts of the prior /
// posterior mixture log-probs in log2 units. Caller applies:
//   prior = acc*ln2 + count*CPRIOR ;  post = acc*ln2 + count*CPOST
__device__ __forceinline__ float sample_and_logp(float pi, float rho1, float rho2,
                                                 float eps, float g0, float g1,
                                                 float& prior_l2, float& post_l2) {
    float s1 = softplus_fast(rho1);
    float s2 = softplus_fast(rho2);
    // argmax over (pi+g0, (1-pi)+g1); index 1 only on strict greater (jnp ties -> 0)
    float a0 = pi + g0;
    float a1 = (1.0f - pi) + g1;
    float sig = (a1 > a0) ? s1 : s2;
    float w = sig * eps;

    // prior: log2( e^{-w^2/2} + 400*e^{-80000*w^2} )   [k-folded into K1/K2]
    float w2 = w * w;
    float e1 = exp2_raw(K1_F * w2);
    float e2 = exp2_raw(K2_F * w2);
    prior_l2 += log2_raw(fmaf(400.0f, e2, e1));

    // posterior: log2( (pi/s1)*e^{-z1^2/2} + ((1-pi)/s2)*e^{-z2^2/2} )
    float rs1 = rcp_fast(s1), rs2 = rcp_fast(s2);
    float z1 = w * rs1, z2 = w * rs2;
    float q1 = exp2_raw(K1_F * z1 * z1);
    float q2 = exp2_raw(K1_F * z2 * z2);
    float c1 = pi * rs1;
    float c2 = fmaf(-pi, rs2, rs2);          // (1-pi)/s2
    post_l2 += log2_raw(fmaf(c2, q2, c1 * q1));
    return w;
}

// ---------------------------------------------------------------------------
// Kernel 1: bias sampling + its log-prob partial sums (8192 elements)
// ---------------------------------------------------------------------------
__global__ __launch_bounds__(256) void bias_kernel(
    const float* __restrict__ b_pi, const float* __restrict__ b_rho1,
    const float* __restrict__ b_rho2, const float* __restrict__ eps_b,
    const float* __restrict__ gum_b,
    float* __restrict__ bias_ws, float* __restrict__ ws_prior, float* __restrict__ ws_post) {
    float prior = 0.0f, post = 0.0f;
    for (int i = threadIdx.x; i < OUTN; i += 256) {           // exactly 32 per thread
        float w = sample_and_logp(b_pi[i], b_rho1[i], b_rho2[i], eps_b[i],
                                  gum_b[2 * i], gum_b[2 * i + 1], prior, post);
        bias_ws[i] = w;
    }
    prior = fmaf(prior, LN2_F, 32.0f * CPRIOR_F);
    post  = fmaf(post,  LN2_F, 32.0f * CPOST_F);

    __shared__ float sp[256], ss[256];
    sp[threadIdx.x] = prior; ss[threadIdx.x] = post;
    __syncthreads();
    for (int s = 128; s > 0; s >>= 1) {
        if (threadIdx.x < (unsigned)s) {
            sp[threadIdx.x] += sp[threadIdx.x + s];
            ss[threadIdx.x] += ss[threadIdx.x + s];
        }
        __syncthreads();
    }
    if (threadIdx.x == 0) { ws_prior[NWG] = sp[0]; ws_post[NWG] = ss[0]; }
}

// ---------------------------------------------------------------------------
// Kernel 2: fused weight sampling + log-prob reduction + GEMM via
// V_WMMA_F32_16X16X4_F32. One wave32 per workgroup; wave owns 16 out-columns
// (weight rows) and all 128 batch rows (8 M-tiles of 16).
//
// Fragment layouts (CDNA5 ISA §7.12.2, 32-bit 16x4 A / 4x16 B / 16x16 C):
//   A vgpr j, lane L: M = L&15,  K = 2*(L>>4) + j
//   B vgpr j, lane L: N = L&15,  K = 2*(L>>4) + j   (mirror of A)
//   C vgpr r, lane L: N = L&15,  M = r + 8*(L>>4)
// ---------------------------------------------------------------------------
__global__ __launch_bounds__(32) void fused_kernel(
    const float* __restrict__ x,
    const float* __restrict__ w_pi, const float* __restrict__ w_rho1,
    const float* __restrict__ w_rho2, const float* __restrict__ eps_w,
    const float* __restrict__ gum_w,
    const float* __restrict__ bias_ws,
    float* __restrict__ out,
    float* __restrict__ ws_prior, float* __restrict__ ws_post) {
    const int wg  = blockIdx.x;       // 0..511
    const int n0  = wg * 16;
    const int L   = threadIdx.x;      // 0..31
    const int row = L & 15;           // N within tile / M within tile for A
    const int kq  = L >> 4;           // 0 or 1
    const int n   = n0 + row;         // this lane's weight row (out column)
    const int wbase = n * INN;        // fits in int: < 2^25

    v8f c[8];
#pragma unroll
    for (int mt = 0; mt < 8; ++mt)
#pragma unroll
        for (int r = 0; r < 8; ++r) c[mt][r] = 0.0f;

    float prior = 0.0f, post = 0.0f;

    for (int k = 0; k < INN; k += 4) {
        const int k0 = k + 2 * kq;

        // prefetch weight streams ~2 cachelines ahead, once per 8 chunks
        if ((k & 31) == 0) {
            __builtin_prefetch(&w_pi  [wbase + k0 + 64], 0, 3);
            __builtin_prefetch(&w_rho1[wbase + k0 + 64], 0, 3);
            __builtin_prefetch(&w_rho2[wbase + k0 + 64], 0, 3);
            __builtin_prefetch(&eps_w [wbase + k0 + 64], 0, 3);
            __builtin_prefetch(&gum_w [(wbase + k0 + 64) * 2], 0, 3);
        }

        const float2 pi2 = *reinterpret_cast<const float2*>(&w_pi  [wbase + k0]);
        const float2 r12 = *reinterpret_cast<const float2*>(&w_rho1[wbase + k0]);
        const float2 r22 = *reinterpret_cast<const float2*>(&w_rho2[wbase + k0]);
        const float2 ep2 = *reinterpret_cast<const float2*>(&eps_w [wbase + k0]);
        const float4 g4  = *reinterpret_cast<const float4*>(&gum_w [(wbase + k0) * 2]);

        v2f b;
        b[0] = sample_and_logp(pi2.x, r12.x, r22.x, ep2.x, g4.x, g4.y, prior, post);
        b[1] = sample_and_logp(pi2.y, r12.y, r22.y, ep2.y, g4.z, g4.w, prior, post);

#pragma unroll
        for (int mt = 0; mt < 8; ++mt) {
            const int m = mt * 16 + row;
            const float2 a2 = *reinterpret_cast<const float2*>(&x[m * INN + k0]);
            v2f a; a[0] = a2.x; a[1] = a2.y;
            c[mt] = __builtin_amdgcn_wmma_f32_16x16x4_f32(
                false, a, false, b, (short)0, c[mt], false, false);
        }
    }

    // apply log2->ln scale + per-element constants (2048 elements per thread)
    prior = fmaf(prior, LN2_F, 2048.0f * CPRIOR_F);
    post  = fmaf(post,  LN2_F, 2048.0f * CPOST_F);

    // add bias and store: C vgpr r, lane L -> (M = mt*16 + r + 8*kq, N = n)
    const float bv = bias_ws[n];
#pragma unroll
    for (int mt = 0; mt < 8; ++mt) {
#pragma unroll
        for (int r = 0; r < 8; ++r) {
            const int m = mt * 16 + r + 8 * kq;
            out[m * OUTN + n] = c[mt][r] + bv;
        }
    }

    // wave-level reduce of the two log-sums, one partial pair per workgroup
    for (int off = 16; off > 0; off >>= 1) {
        prior += __shfl_down(prior, off, 32);
        post  += __shfl_down(post,  off, 32);
    }
    if (L == 0) { ws_prior[wg] = prior; ws_post[wg] = post; }
}

// ---------------------------------------------------------------------------
// Kernel 3: deterministic fixed-order final reduction of 513 partial pairs
// ---------------------------------------------------------------------------
__global__ void finalize_kernel(const float* __restrict__ ws_prior,
                                const float* __restrict__ ws_post,
                                float* __restrict__ scalars) {
    if (threadIdx.x == 0 && blockIdx.x == 0) {
        double p = 0.0, q = 0.0;
        for (int i = 0; i <= NWG; ++i) { p += (double)ws_prior[i]; q += (double)ws_post[i]; }
        scalars[0] = (float)p;   // log_prior
        scalars[1] = (float)q;   // log_post
    }
}

extern "C" void kernel_launch(void* const* d_in, const int* in_sizes, int n_in,
                              void* d_out, int out_size, void* d_ws, size_t ws_size,
                              hipStream_t stream) {
    const float* x      = (const float*)d_in[0];
    const float* w_pi   = (const float*)d_in[1];
    const float* w_rho1 = (const float*)d_in[2];
    const float* w_rho2 = (const float*)d_in[3];
    const float* b_pi   = (const float*)d_in[4];
    const float* b_rho1 = (const float*)d_in[5];
    const float* b_rho2 = (const float*)d_in[6];
    const float* eps_w  = (const float*)d_in[7];
    const float* eps_b  = (const float*)d_in[8];
    const float* gum_w  = (const float*)d_in[9];
    const float* gum_b  = (const float*)d_in[10];

    float* out     = (float*)d_out;                     // [128, 8192]
    float* scalars = out + (size_t)BATCH * OUTN;        // log_prior, log_post

    float* ws_bias  = (float*)d_ws;                     // 8192 floats
    float* ws_prior = ws_bias + OUTN;                   // NWG+1 floats
    float* ws_post  = ws_prior + (NWG + 64);            // NWG+1 floats (padded)

    bias_kernel<<<1, 256, 0, stream>>>(b_pi, b_rho1, b_rho2, eps_b, gum_b,
                                       ws_bias, ws_prior, ws_post);
    fused_kernel<<<NWG, 32, 0, stream>>>(x, w_pi, w_rho1, w_rho2, eps_w, gum_w,
                                         ws_bias, out, ws_prior, ws_post);
    finalize_kernel<<<1, 32, 0, stream>>>(ws_prior, ws_post, scalars);
}